// ProbAttention_33406255628788
// MI455X (gfx1250) — compile-verified
//
#include <hip/hip_runtime.h>
#include <math.h>

// ProbSparse attention (Informer) for MI455X / gfx1250, fp32 end-to-end.
// B=4 H=8 LQ=LK=2048 D=64, SAMPLE_K=N_TOP=40, SCALE=1/8.

typedef __attribute__((ext_vector_type(2))) float v2f;
typedef __attribute__((ext_vector_type(8))) float v8f;

#define B_      4
#define H_      8
#define LQ_     2048
#define LK_     2048
#define D_      64
#define SK_     40            // SAMPLE_K
#define NT_     40            // N_TOP
#define SCALE_  0.125f
#define NKT_    (LK_ / 16)    // 128 key tiles of 16
#define CHUNK_  128
#define NCHUNK_ (LK_ / CHUNK_) // 16

// ---------------------------------------------------------------------------
// fp32 WMMA: D(16x16) = A(16x4) * B(4x16) + C, exact fp32 accumulate.
// A layout (ISA 7.12.2): lane<16 -> M=lane, regs=(K0,K1); lane>=16 -> (K2,K3).
// B layout (by symmetry): lane&15 = N, half selects K-pair {0,1} / {2,3}.
// C/D layout: VGPR r -> M = r + 8*(lane>=16), N = lane&15.
// ---------------------------------------------------------------------------
__device__ __forceinline__ v8f wmma_f32(v2f a, v2f b, v8f c) {
  return __builtin_amdgcn_wmma_f32_16x16x4_f32(
      /*neg_a=*/false, a, /*neg_b=*/false, b,
      /*c_mod=*/(short)0, c, /*reuse_a=*/false, /*reuse_b=*/false);
}

// reductions within each 16-lane half (xor masks 1..8 never cross halves)
__device__ __forceinline__ float grp16_max(float v) {
#pragma unroll
  for (int m = 1; m < 16; m <<= 1) v = fmaxf(v, __shfl_xor(v, m, 32));
  return v;
}
__device__ __forceinline__ float grp16_sum(float v) {
#pragma unroll
  for (int m = 1; m < 16; m <<= 1) v += __shfl_xor(v, m, 32);
  return v;
}

// ---------------------------------------------------------------------------
// Stage 1: M[bh,l] = max_s(q.k_s) - mean_s(q.k_s) over 40 sampled keys.
// One wave per query: lane holds dims 2j,2j+1; coalesced float2 loads of the
// gathered K row (L2-resident), full-wave shfl reduction per sample.
// ---------------------------------------------------------------------------
__global__ void __launch_bounds__(256) pa_mscore(
    const float* __restrict__ Q, const float* __restrict__ K,
    const int* __restrict__ isamp, float* __restrict__ Mbuf) {
  const int bh = blockIdx.x;
  const int b = bh / H_, h = bh % H_;
  const int wave = threadIdx.x >> 5;
  const int lane = threadIdx.x & 31;
  const int l = blockIdx.y * 8 + wave;

  const float2 q = *(const float2*)(Q + (((size_t)b * LQ_ + l) * H_ + h) * D_ + lane * 2);
  const float* Kb = K + ((size_t)b * LK_ * H_ + h) * D_;

  float mx = -INFINITY, sm = 0.f;
  for (int s = 0; s < SK_; ++s) {
    const int key = isamp[l * SK_ + s];
    const float2 kv = *(const float2*)(Kb + (size_t)key * (H_ * D_) + lane * 2);
    float part = q.x * kv.x + q.y * kv.y;
#pragma unroll
    for (int m = 1; m < 32; m <<= 1) part += __shfl_xor(part, m, 32);
    mx = fmaxf(mx, part);
    sm += part;
  }
  if (lane == 0) Mbuf[(size_t)bh * LQ_ + l] = mx - sm * (1.f / SK_);
}

// ---------------------------------------------------------------------------
// Stage 2: top-40 per (b,h) by iterative argmax over LDS-resident M.
// Tie-break to the smallest index (matches top_k first-occurrence).
// ---------------------------------------------------------------------------
__global__ void __launch_bounds__(256) pa_topk(
    const float* __restrict__ Mbuf, int* __restrict__ idxbuf) {
  const int bh = blockIdx.x;
  const int t = threadIdx.x;
  __shared__ float mv[LQ_];
  __shared__ float rv[256];
  __shared__ int ri[256];

  for (int i = t; i < LQ_; i += 256) mv[i] = Mbuf[(size_t)bh * LQ_ + i];
  __syncthreads();

  for (int iter = 0; iter < NT_; ++iter) {
    float best = -INFINITY;
    int bidx = 0x7fffffff;
    for (int i = t; i < LQ_; i += 256) {  // ascending -> strict > keeps first
      const float v = mv[i];
      if (v > best) { best = v; bidx = i; }
    }
    rv[t] = best; ri[t] = bidx;
    __syncthreads();
    for (int off = 128; off > 0; off >>= 1) {
      if (t < off) {
        const float ov = rv[t + off]; const int oi = ri[t + off];
        if (ov > rv[t] || (ov == rv[t] && oi < ri[t])) { rv[t] = ov; ri[t] = oi; }
      }
      __syncthreads();
    }
    if (t == 0) { idxbuf[bh * 48 + iter] = ri[0]; mv[ri[0]] = -INFINITY; }
    __syncthreads();
  }
  if (t < 8) idxbuf[bh * 48 + NT_ + t] = 0;  // pad rows 40..47
}

// ---------------------------------------------------------------------------
// Stage 3: context = cumsum(V) along L for every row (two-phase chunk scan).
// ---------------------------------------------------------------------------
__global__ void __launch_bounds__(64) pa_csum1(
    const float* __restrict__ V, float* __restrict__ csums) {
  const int bh = blockIdx.x, chunk = blockIdx.y, d = threadIdx.x;
  const int b = bh / H_, h = bh % H_;
  const float* Vb = V + ((size_t)b * LK_ * H_ + h) * D_ + d;
  const int l0 = chunk * CHUNK_;
  float s = 0.f;
  for (int i = 0; i < CHUNK_; ++i) s += Vb[(size_t)(l0 + i) * (H_ * D_)];
  csums[((size_t)bh * NCHUNK_ + chunk) * D_ + d] = s;
}

__global__ void __launch_bounds__(64) pa_csum2(
    const float* __restrict__ V, const float* __restrict__ csums,
    float* __restrict__ out) {
  const int bh = blockIdx.x, chunk = blockIdx.y, d = threadIdx.x;
  const int b = bh / H_, h = bh % H_;
  float run = 0.f;
  for (int c = 0; c < chunk; ++c) run += csums[((size_t)bh * NCHUNK_ + c) * D_ + d];
  const float* Vb = V + ((size_t)b * LK_ * H_ + h) * D_ + d;
  float* Ob = out + ((size_t)b * LQ_ * H_ + h) * D_ + d;
  const int l0 = chunk * CHUNK_;
  for (int i = 0; i < CHUNK_; ++i) {
    run += Vb[(size_t)(l0 + i) * (H_ * D_)];
    Ob[(size_t)(l0 + i) * (H_ * D_)] = run;  // top rows overwritten by pa_flash
  }
}

// ---------------------------------------------------------------------------
// Stage 4: two-pass flash attention of the 40 selected queries vs all keys.
// One block per (b,h); 3 waves, each owns a 16-query M-tile (40 padded to 48).
// Pass 1: per-lane running max via WMMA, ONE cross-lane max at the end.
// Pass 2: recompute S, P=exp(SCALE*(S-max)), per-lane running sum, acc += P@V
// via WMMA (P transposed D->A layout through wave-private LDS); ONE cross-lane
// sum before the epilogue. Keeps the hot loops free of ds_bpermute chains.
// ---------------------------------------------------------------------------
__global__ void __launch_bounds__(96) pa_flash(
    const float* __restrict__ Q, const float* __restrict__ K,
    const float* __restrict__ V, const int* __restrict__ idxbuf,
    float* __restrict__ out) {
  const int bh = blockIdx.x;
  const int b = bh / H_, h = bh % H_;
  const int wave = threadIdx.x >> 5;   // M tile 0..2
  const int lane = threadIdx.x & 31;
  const int n16 = lane & 15;
  const int khalf = lane >> 4;         // 0: K{0,1}, 1: K{2,3}

  __shared__ float plds[3][16][17];    // wave-private P transpose staging

  // A fragments: 16 queries x 64 dims = 16 k-steps of 16x16x4.
  const int u = wave * 16 + n16;
  int lq = 0; float qmask = 0.f;
  if (u < NT_) { lq = idxbuf[bh * 48 + u]; qmask = 1.f; }  // pad rows -> A=0

  const float* qrow = Q + (((size_t)b * LQ_ + lq) * H_ + h) * D_;
  v2f afrag[16];
#pragma unroll
  for (int kk = 0; kk < 16; ++kk) {
    const int d0 = kk * 4 + khalf * 2;
    const float2 t = *(const float2*)(qrow + d0);
    afrag[kk].x = t.x * qmask;
    afrag[kk].y = t.y * qmask;
  }

  const float* Kb = K + ((size_t)b * LK_ * H_ + h) * D_;
  const float* Vb = V + ((size_t)b * LK_ * H_ + h) * D_;

  // ---- pass 1: per-lane running max of raw scores (reduce once at end) ----
  float rowmax[8];
#pragma unroll
  for (int r = 0; r < 8; ++r) rowmax[r] = -INFINITY;

  for (int kt = 0; kt < NKT_; ++kt) {
    const float* krow = Kb + (size_t)(kt * 16 + n16) * (H_ * D_);
    v8f s = {};
#pragma unroll
    for (int kk = 0; kk < 16; ++kk) {
      const int d0 = kk * 4 + khalf * 2;
      const float2 t = *(const float2*)(krow + d0);
      v2f bf; bf.x = t.x; bf.y = t.y;
      s = wmma_f32(afrag[kk], bf, s);
    }
#pragma unroll
    for (int r = 0; r < 8; ++r) rowmax[r] = fmaxf(rowmax[r], s[r]);
  }
  // deferred cross-lane reduction: true row max, replicated across the half
#pragma unroll
  for (int r = 0; r < 8; ++r) rowmax[r] = grp16_max(rowmax[r]);

  // ---- pass 2: unnormalized softmax @ V, per-lane running row sums ----
  float rowsum[8];
#pragma unroll
  for (int r = 0; r < 8; ++r) rowsum[r] = 0.f;
  const v8f vzero = {};
  v8f acc[4];
#pragma unroll
  for (int nt = 0; nt < 4; ++nt) acc[nt] = vzero;

  for (int kt = 0; kt < NKT_; ++kt) {
    const float* krow = Kb + (size_t)(kt * 16 + n16) * (H_ * D_);
    v8f s = {};
#pragma unroll
    for (int kk = 0; kk < 16; ++kk) {
      const int d0 = kk * 4 + khalf * 2;
      const float2 t = *(const float2*)(krow + d0);
      v2f bf; bf.x = t.x; bf.y = t.y;
      s = wmma_f32(afrag[kk], bf, s);
    }
    // P = exp(SCALE*(S - rowmax)); s[r] is row (wave*16 + r + 8*khalf)
#pragma unroll
    for (int r = 0; r < 8; ++r) {
      const float p = __expf((s[r] - rowmax[r]) * SCALE_);
      rowsum[r] += p;                       // lane-local; reduced after loop
      plds[wave][r + 8 * khalf][n16] = p;   // D-layout -> LDS [row][col]
    }
    // P (A-layout) @ V tile: K = 16 keys -> 4 k-steps; N = 64 dims -> 4 tiles
#pragma unroll
    for (int kk2 = 0; kk2 < 4; ++kk2) {
      const int k0 = kk2 * 4 + khalf * 2;
      v2f pa;
      pa.x = plds[wave][n16][k0];           // wave-private: in-order DS, no barrier
      pa.y = plds[wave][n16][k0 + 1];
      const float* vrow0 = Vb + (size_t)(kt * 16 + k0) * (H_ * D_);
      const float* vrow1 = vrow0 + H_ * D_;
#pragma unroll
      for (int nt = 0; nt < 4; ++nt) {
        v2f vb;
        vb.x = vrow0[nt * 16 + n16];
        vb.y = vrow1[nt * 16 + n16];
        acc[nt] = wmma_f32(pa, vb, acc[nt]);
      }
    }
  }
  // deferred cross-lane reduction: true row sums
#pragma unroll
  for (int r = 0; r < 8; ++r) rowsum[r] = grp16_sum(rowsum[r]);

  // ---- epilogue: scatter normalized rows onto out[b, idx[u], h, :] ----
#pragma unroll
  for (int r = 0; r < 8; ++r) {
    const int row = wave * 16 + r + 8 * khalf;
    if (row < NT_) {
      const int l = idxbuf[bh * 48 + row];
      float* orow = out + (((size_t)b * LQ_ + l) * H_ + h) * D_;
      const float inv = 1.f / rowsum[r];
#pragma unroll
      for (int nt = 0; nt < 4; ++nt) orow[nt * 16 + n16] = acc[nt][r] * inv;
    }
  }
}

// ---------------------------------------------------------------------------
extern "C" void kernel_launch(void* const* d_in, const int* in_sizes, int n_in,
                              void* d_out, int out_size, void* d_ws, size_t ws_size,
                              hipStream_t stream) {
  (void)in_sizes; (void)n_in; (void)out_size; (void)ws_size;
  const float* Q = (const float*)d_in[0];      // (B, LQ, H, D) f32
  const float* K = (const float*)d_in[1];      // (B, LK, H, D) f32
  const float* V = (const float*)d_in[2];      // (B, LK, H, D) f32
  const int* isamp = (const int*)d_in[3];      // (LQ, SAMPLE_K) i32
  float* out = (float*)d_out;                  // (B, LQ, H, D) f32

  // workspace: M scores (256 KB) | top-k indices (6 KB) | chunk sums (128 KB)
  float* Mbuf = (float*)d_ws;
  int* idxb = (int*)((char*)d_ws + (size_t)B_ * H_ * LQ_ * sizeof(float));
  float* csums = (float*)((char*)idxb + (size_t)B_ * H_ * 48 * sizeof(int));

  pa_mscore<<<dim3(B_ * H_, LQ_ / 8), 256, 0, stream>>>(Q, K, isamp, Mbuf);
  pa_topk<<<dim3(B_ * H_), 256, 0, stream>>>(Mbuf, idxb);
  pa_csum1<<<dim3(B_ * H_, NCHUNK_), D_, 0, stream>>>(V, csums);
  pa_csum2<<<dim3(B_ * H_, NCHUNK_), D_, 0, stream>>>(V, csums, out);
  pa_flash<<<dim3(B_ * H_), 96, 0, stream>>>(Q, K, V, idxb, out);
}